// SafeCTCLoss_9680856285412
// MI455X (gfx1250) — compile-verified
//
#include <hip/hip_runtime.h>
#include <stdint.h>

// ---------------- problem constants (from reference) ----------------
#define T_LEN 1024
#define B_SZ  256
#define C_CLS 512
#define S_TGT 128
#define L_EXT (2 * S_TGT + 1)   // 257 extended labels
#define NEGF  (-1e30f)

// ---------------- kernel tuning ----------------
#define DEPTH    8               // LDS row ring buffers (2KB each)
#define NTHREADS 288             // 9 wave32: 257 alpha lanes + round up
#define LOADERS  128             // 128 lanes x b128 = 2048B = one [C] row

#define LOG2E 1.4426950408889634f
#define LN2   0.6931471805599453f
#define NEGL  (NEGF * LOG2E)     // NEG in log2 domain (finite f32)

// ---------------- CDNA5 fast transcendentals (v_exp_f32 / v_log_f32) ----
__device__ __forceinline__ float fast_exp2(float x) {
#if __has_builtin(__builtin_amdgcn_exp2f)
  return __builtin_amdgcn_exp2f(x);
#else
  return exp2f(x);
#endif
}
__device__ __forceinline__ float fast_log2(float x) {
#if __has_builtin(__builtin_amdgcn_logf)
  return __builtin_amdgcn_logf(x);   // base-2 log (v_log_f32)
#else
  return log2f(x);
#endif
}

// ---------------- CDNA5 async global->LDS DMA path ----------------
// VDST supplies the LDS byte address; generic (flat) LDS pointers carry the LDS
// byte offset in their low 32 bits (ISA 10.2 aperture rules), so casting a
// __shared__ address to u32 yields a valid VDST value.
__device__ __forceinline__ void async_copy_b128(uint32_t lds_byte_off,
                                                uint64_t gaddr) {
#if defined(__HIP_DEVICE_COMPILE__)
  asm volatile("global_load_async_to_lds_b128 %0, %1, off"
               :
               : "v"(lds_byte_off), "v"(gaddr)
               : "memory");
#endif
}

// Inline-asm only: the "memory" clobber is a compiler fence, guaranteeing the
// wait cannot be reordered across the workgroup barrier (the cross-wave DMA
// protocol requires wait-before-barrier ordering).
template <int N>
__device__ __forceinline__ void wait_asynccnt() {
#if defined(__HIP_DEVICE_COMPILE__)
  asm volatile("s_wait_asynccnt %0" : : "n"(N) : "memory");
#endif
}

// =====================================================================
// Kernel 1: one workgroup per batch. Alpha recursion over T with an
// async-DMA-fed LDS ring of log-prob rows and LDS neighbor exchange.
// =====================================================================
__global__ __launch_bounds__(NTHREADS) void ctc_alpha_kernel(
    const float* __restrict__ logp,     // [T, B, C]
    const int*   __restrict__ targets,  // [B, S]
    const int*   __restrict__ inlen,    // [B]
    const int*   __restrict__ tlen,     // [B]
    float*       __restrict__ loss_out) // [B] raw per-sample loss
{
  __shared__ __align__(16) float rows[DEPTH][C_CLS];  // 16 KB DMA ring
  __shared__ float ash[2][L_EXT + 2];                 // alpha ping-pong (+2 NEG pads)

  const int b   = blockIdx.x;
  const int tid = threadIdx.x;

  // ---- per-thread extended-label setup: ext[s], allow_skip[s] ----
  // ext[s] = blank(0) for even s, targets[s/2] for odd s.
  int  ext     = 0;
  bool canskip = false;
  if (tid < L_EXT && (tid & 1)) {
    const int* tg = targets + (size_t)b * S_TGT;
    const int lab = tg[tid >> 1];
    ext = lab;
    if (tid >= 3) canskip = (lab != tg[(tid >> 1) - 1]);
  }
  const int il = inlen[b];

  if (tid < 2) { ash[0][tid] = NEGL; ash[1][tid] = NEGL; }  // pad slots

  // ---- DMA setup ----
  const uint64_t rowbytes = (uint64_t)B_SZ * C_CLS * 4ull;  // t-stride
  const uint64_t gbase =
      (uint64_t)(uintptr_t)logp + (uint64_t)b * C_CLS * 4ull + (uint64_t)(tid * 16);
  const uint32_t lds_row0 =
      (uint32_t)(uintptr_t)(&rows[0][0]) + (uint32_t)(tid * 16);
  const bool loader = (tid < LOADERS);   // waves 0..3, wave-uniform

  // Prologue: issue rows 0 .. DEPTH-2 (prefetch distance DEPTH-2).
  if (loader) {
#pragma unroll
    for (int r = 0; r < DEPTH - 1; ++r) {
      async_copy_b128(lds_row0 + (uint32_t)(r * C_CLS * 4),
                      gbase + (uint64_t)r * rowbytes);
    }
  }

  // ---- t = 0 init ----
  wait_asynccnt<DEPTH - 2>();   // row 0 resident (in-order completion)
  __syncthreads();
  float A = NEGL;               // alpha in log2 domain
  if (tid < L_EXT) {
    const float lp0 = rows[0][ext] * LOG2E;
    A = (tid < 2) ? lp0 : NEGL;
  }

  // ---- main recursion, steady state: t = 1 .. T-DEPTH+1 -----------------
  // At the wait point of iter t, issued rows reach t+DEPTH-3, so pending
  // <= DEPTH-3 guarantees row t resident (async loads complete in order).
  // Prefetch row t+DEPTH-2 lands in the buffer consumed at iter t-2, whose
  // reads finished before the barrier of iter t-1 -> race-free with a single
  // barrier per iteration (alpha LDS is ping-ponged for the same reason).
  int t = 1;
  for (; t <= T_LEN - DEPTH + 1; ++t) {
    const int p = (t + 1) & 1;                 // == (t-1)&1 : alpha buffer
    if (tid < L_EXT) ash[p][tid + 2] = A;      // publish alpha[t-1]

    wait_asynccnt<DEPTH - 3>();
    __syncthreads();                           // row t + alpha visible to all

    const int tn = t + DEPTH - 2;              // always < T_LEN in this loop
    if (loader) {
      async_copy_b128(lds_row0 + (uint32_t)((tn % DEPTH) * C_CLS * 4),
                      gbase + (uint64_t)tn * rowbytes);
    }

    if (tid < L_EXT) {
      const float lp = rows[t % DEPTH][ext] * LOG2E;  // LDS gather
      const float a1 = ash[p][tid + 1];               // alpha[s-1]
      const float a2 = canskip ? ash[p][tid] : NEGL;  // alpha[s-2]
      const float m  = fmaxf(A, fmaxf(a1, a2));
      const float s  = fast_exp2(A - m) + fast_exp2(a1 - m) + fast_exp2(a2 - m);
      const float nv = m + fast_log2(s) + lp;
      A = (t < il) ? nv : A;                          // freeze past input_len
    }
  }

  // ---- drain tail: all rows already issued; wait for everything ---------
  for (; t < T_LEN; ++t) {
    const int p = (t + 1) & 1;
    if (tid < L_EXT) ash[p][tid + 2] = A;

    wait_asynccnt<0>();
    __syncthreads();

    if (tid < L_EXT) {
      const float lp = rows[t % DEPTH][ext] * LOG2E;
      const float a1 = ash[p][tid + 1];
      const float a2 = canskip ? ash[p][tid] : NEGL;
      const float m  = fmaxf(A, fmaxf(a1, a2));
      const float s  = fast_exp2(A - m) + fast_exp2(a1 - m) + fast_exp2(a2 - m);
      const float nv = m + fast_log2(s) + lp;
      A = (t < il) ? nv : A;
    }
  }

  // ---- per-sample loss: -logaddexp(alpha[2L], alpha[2L-1]) ----
  __syncthreads();
  if (tid < L_EXT) ash[0][tid + 2] = A;
  __syncthreads();
  if (tid == 0) {
    const int   tl  = tlen[b];
    const int   idx = 2 * tl;                 // final blank position (<= L-1)
    const float ae  = ash[0][idx + 2];
    const float ae2 = ash[0][idx + 1];        // idx-1 (pad NEG if idx==0)
    const float m   = fmaxf(ae, ae2);
    const float l2  = m + fast_log2(fast_exp2(ae - m) + fast_exp2(ae2 - m));
    loss_out[b] = -(l2 * LN2);                // back to natural log
  }
}

// =====================================================================
// Kernel 2: deterministic single-block reduction -> mean over valid.
// valid = (tl <= il) & (il > 0) & (global_max(targets) <= C)
// loss  = where(isfinite & valid, loss, 0); out = sum(loss)/max(nvalid,1)
// =====================================================================
__global__ __launch_bounds__(256) void ctc_reduce_kernel(
    const float* __restrict__ loss_in,
    const int*   __restrict__ targets,
    const int*   __restrict__ inlen,
    const int*   __restrict__ tlen,
    float*       __restrict__ out)
{
  __shared__ int   smax[256];
  __shared__ float ssum[256];
  __shared__ float scnt[256];
  const int tid = threadIdx.x;

  // global max over all targets (scalar condition in the reference)
  int gm = 0;
  for (int i = tid; i < B_SZ * S_TGT; i += 256) gm = max(gm, targets[i]);
  smax[tid] = gm;
  __syncthreads();
  for (int off = 128; off > 0; off >>= 1) {
    if (tid < off) smax[tid] = max(smax[tid], smax[tid + off]);
    __syncthreads();
  }
  const bool tgt_ok = (smax[0] <= C_CLS);

  float lsum = 0.0f, lcnt = 0.0f;
  for (int i = tid; i < B_SZ; i += 256) {
    const int  tl    = tlen[i];
    const int  ilen  = inlen[i];
    const bool valid = (tl <= ilen) && (ilen > 0) && tgt_ok;
    float l = loss_in[i];
    if (!(l * 0.0f == 0.0f)) l = 0.0f;   // finite <=> l*0 == 0 (kills NaN/Inf)
    if (!valid) l = 0.0f;
    lsum += l;
    lcnt += valid ? 1.0f : 0.0f;
  }
  ssum[tid] = lsum;
  scnt[tid] = lcnt;
  __syncthreads();
  for (int off = 128; off > 0; off >>= 1) {
    if (tid < off) { ssum[tid] += ssum[tid + off]; scnt[tid] += scnt[tid + off]; }
    __syncthreads();
  }
  if (tid == 0) out[0] = ssum[0] / fmaxf(scnt[0], 1.0f);
}

// =====================================================================
extern "C" void kernel_launch(void* const* d_in, const int* in_sizes, int n_in,
                              void* d_out, int out_size, void* d_ws, size_t ws_size,
                              hipStream_t stream) {
  const float* logp    = (const float*)d_in[0];  // [T,B,C] f32
  const int*   targets = (const int*)d_in[1];    // [B,S]   i32
  const int*   inlen   = (const int*)d_in[2];    // [B]     i32
  const int*   tlen    = (const int*)d_in[3];    // [B]     i32
  float*       loss_ws = (float*)d_ws;           // B floats scratch
  float*       outp    = (float*)d_out;          // scalar mean loss

  ctc_alpha_kernel<<<B_SZ, NTHREADS, 0, stream>>>(logp, targets, inlen, tlen, loss_ws);
  ctc_reduce_kernel<<<1, 256, 0, stream>>>(loss_ws, targets, inlen, tlen, outp);
}